// CEMA_34445637714419
// MI455X (gfx1250) — compile-verified
//
#include <hip/hip_runtime.h>
#include <math.h>

typedef __attribute__((ext_vector_type(2))) float v2f;
typedef __attribute__((ext_vector_type(4))) float v4f;
typedef __attribute__((ext_vector_type(8))) float v8f;

#define DDIM  2048
#define HDIM  64
#define BATCH 16384

// ---------------------------------------------------------------------------
// Kernel 1: scale[d] = eta[d,:] @ C @ (alpha*beta)[d,:],
//           C[j][k] = cos(j * omega[k] * 2*pi/64)
// Block = 128 threads (4 wave32), handles 64 consecutive d-rows.
// Each wave computes T = eta_tile(16x64) @ C(64x64) via V_WMMA_F32_16X16X4_F32
// (16 K-steps per 16-wide N-tile, 4 N-tiles), then dots T against alpha*beta.
// ---------------------------------------------------------------------------
__global__ __launch_bounds__(128) void cema_scale_kernel(
    const float* __restrict__ alpha, const float* __restrict__ omega,
    const float* __restrict__ beta,  const float* __restrict__ eta,
    float* __restrict__ scale)
{
    __shared__ float sC [HDIM * HDIM];   // cosine matrix, 16 KB
    __shared__ float sAB[64 * HDIM];     // alpha*beta tile, 16 KB
    __shared__ float sE [64 * HDIM];     // eta tile,        16 KB

    const int tid = threadIdx.x;
    const int d0  = blockIdx.x * 64;
    const float w0 = 6.28318530717958647692f / (float)HDIM;

    for (int idx = tid; idx < HDIM * HDIM; idx += 128) {
        int j = idx >> 6, k = idx & 63;
        sC[idx] = __cosf((float)j * omega[k] * w0);
    }
    for (int idx = tid; idx < 64 * HDIM; idx += 128) {
        int g = d0 * HDIM + idx;          // global (d,k) flat index
        sAB[idx] = alpha[g] * beta[g];
        sE[idx]  = eta[g];
    }
    __syncthreads();

    const int wave = tid >> 5;
    const int lane = tid & 31;
    const int half = lane >> 4;           // 0: lanes 0-15, 1: lanes 16-31
    const int ln   = lane & 15;
    const int rowA = wave * 16 + ln;      // A-fragment row (M = lane mod 16)
    const int rowD = wave * 16 + half*8;  // first D-fragment row in this half

    float s[8];
    #pragma unroll
    for (int r = 0; r < 8; ++r) s[r] = 0.f;

    #pragma unroll
    for (int nn = 0; nn < 4; ++nn) {      // N-tiles of C (columns k)
        v8f c = {};
        #pragma unroll
        for (int kk = 0; kk < 16; ++kk) { // K-steps of 4 (index j)
            const int kbase = kk * 4 + half * 2;
            v2f a, b;
            // A 16x4 f32 layout: lanes0-15 hold K0/K1, lanes16-31 hold K2/K3
            a.x = sE[rowA * HDIM + kbase + 0];
            a.y = sE[rowA * HDIM + kbase + 1];
            // B 4x16 f32 layout mirrors A: row K striped over lanes (N=ln)
            b.x = sC[(kbase + 0) * HDIM + nn * 16 + ln];
            b.y = sC[(kbase + 1) * HDIM + nn * 16 + ln];
            c = __builtin_amdgcn_wmma_f32_16x16x4_f32(
                    false, a, false, b, (short)0, c, false, false);
        }
        // D layout: VGPR r -> row rowD+r, column nn*16+ln
        #pragma unroll
        for (int r = 0; r < 8; ++r)
            s[r] += c[r] * sAB[(rowD + r) * HDIM + nn * 16 + ln];
    }

    // Reduce over the 16 lanes of each half-wave (columns of T)
    #pragma unroll
    for (int m = 1; m < 16; m <<= 1) {
        #pragma unroll
        for (int r = 0; r < 8; ++r)
            s[r] += __shfl_xor(s[r], m, 32);
    }
    if (ln == 0) {
        #pragma unroll
        for (int r = 0; r < 8; ++r)
            scale[d0 + rowD + r] = s[r];
    }
}

// ---------------------------------------------------------------------------
// Kernel 2: y[b,d] = x[b,d] * scale[d]. Pure HBM-bandwidth stream (268 MB).
// 128-bit vectorized, non-temporal on the streams, cached scale lookups.
// ---------------------------------------------------------------------------
__global__ __launch_bounds__(256) void cema_apply_kernel(
    const v4f* __restrict__ x, const float* __restrict__ scale,
    v4f* __restrict__ y)
{
    const int idx = blockIdx.x * 256 + threadIdx.x;   // exact-sized grid
    const int d4  = idx & (DDIM / 4 - 1);             // D divides vec width
    v4f xs = __builtin_nontemporal_load(&x[idx]);
    v4f sc = ((const v4f* __restrict__)scale)[d4];    // 8 KB table: stays hot
    v4f out = xs * sc;
    __builtin_nontemporal_store(out, &y[idx]);
}

extern "C" void kernel_launch(void* const* d_in, const int* in_sizes, int n_in,
                              void* d_out, int out_size, void* d_ws, size_t ws_size,
                              hipStream_t stream) {
    const float* x     = (const float*)d_in[0];
    const float* alpha = (const float*)d_in[1];
    /* d_in[2] = delta, unused by the reference math */
    const float* omega = (const float*)d_in[3];
    const float* beta  = (const float*)d_in[4];
    const float* eta   = (const float*)d_in[5];

    float* scale = (float*)d_ws;        // 2048 floats of scratch
    float* y     = (float*)d_out;

    cema_scale_kernel<<<DDIM / 64, 128, 0, stream>>>(alpha, omega, beta, eta, scale);

    const int total4 = (BATCH * DDIM) / 4;            // 8,388,608
    cema_apply_kernel<<<total4 / 256, 256, 0, stream>>>(
        (const v4f*)x, scale, (v4f*)y);
}